// FocalLoss_51342039057016
// MI455X (gfx1250) — compile-verified
//
#include <hip/hip_runtime.h>
#include <math.h>

// ---------------------------------------------------------------------------
// RetinaNet focal loss for gfx1250 (MI455X).
// Memory-bound: ~325 MB streamed -> ~14 us floor at 23.3 TB/s.
// CDNA5 paths used: global_load_async_to_lds_b128 (annotation staging, tracked
// by ASYNCcnt) + s_wait_asynccnt, b128 vector streaming loads.
// No matrix ops exist in this workload, so no WMMA is emitted (correctly).
// ---------------------------------------------------------------------------

#define BLK   256
#define MAXM  64          // LDS staging sized for up to 64 GT boxes (M=32 here)
#define ALPHA_C 0.25f

#if defined(__has_builtin)
#  if __has_builtin(__builtin_amdgcn_global_load_async_to_lds_b128)
#    define HAVE_ASYNC_B128 1
#  endif
#  if __has_builtin(__builtin_amdgcn_s_wait_asynccnt)
#    define HAVE_WAIT_ASYNC 1
#  endif
#endif

// Matches the builtin's parameter type reported by clang:
//   '__attribute__((__vector_size__(4 * sizeof(int)))) int __device__ *'
typedef int v4i __attribute__((vector_size(16)));
typedef __attribute__((address_space(1))) v4i* v4i_gptr;   // prints as "__device__"
typedef __attribute__((address_space(3))) v4i* v4i_lptr;   // LDS

__device__ __forceinline__ float block_reduce_sum(float v, float* sred, int tid) {
  sred[tid] = v;
  __syncthreads();
#pragma unroll
  for (int off = BLK / 2; off > 0; off >>= 1) {
    if (tid < off) sred[tid] += sred[tid + off];
    __syncthreads();
  }
  float r = sred[0];
  __syncthreads();
  return r;
}

__global__ __launch_bounds__(BLK)
void focal_main(const float* __restrict__ cls,
                const float* __restrict__ regr,
                const float* __restrict__ anch,
                const float* __restrict__ ann,
                float* __restrict__ part,
                int A, int K, int M, int nblk) {
  __shared__ __align__(16) float s_ann[MAXM * 5];
  __shared__ float s_area[MAXM];
  __shared__ float s_valid[MAXM];
  __shared__ float s_red[BLK];

  const int tid = threadIdx.x;
  const int bx  = blockIdx.x;
  const int b   = blockIdx.y;
  const float* ann_img = ann + (size_t)b * M * 5;
  const int annN = M * 5;

  // ---- stage annotations into LDS (async DMA path on gfx1250) ----
  bool did_async = false;
#if defined(HAVE_ASYNC_B128)
  if ((annN & 3) == 0 && annN <= MAXM * 5) {
    const int n16 = annN >> 2;   // number of 16-byte chunks
    if (tid < n16) {
      __builtin_amdgcn_global_load_async_to_lds_b128(
          (v4i_gptr)(float*)(ann_img + tid * 4),
          (v4i_lptr)(&s_ann[tid * 4]),
          0, 0);
    }
#  if defined(HAVE_WAIT_ASYNC)
    __builtin_amdgcn_s_wait_asynccnt(0);
#  else
    asm volatile("s_wait_asynccnt 0" ::: "memory");
#  endif
    did_async = true;
  }
#endif
  if (!did_async) {
    for (int i = tid; i < annN; i += BLK) s_ann[i] = ann_img[i];
  }
  __syncthreads();

  if (tid < M) {
    const float x1 = s_ann[tid * 5 + 0], y1 = s_ann[tid * 5 + 1];
    const float x2 = s_ann[tid * 5 + 2], y2 = s_ann[tid * 5 + 3];
    s_area[tid]  = (x2 - x1) * (y2 - y1);
    s_valid[tid] = (s_ann[tid * 5 + 4] != -1.0f) ? 1.0f : 0.0f;
  }
  __syncthreads();

  float cls_acc = 0.0f, reg_acc = 0.0f, pos_f = 0.0f;
  const int a = bx * BLK + tid;
  if (a < A) {
    const float4 av = *(const float4*)(anch + (size_t)a * 4);
    const float ax1 = av.x, ay1 = av.y, ax2 = av.z, ay2 = av.w;
    const float area_a = (ax2 - ax1) * (ay2 - ay1);

    // ---- IoU argmax over GT boxes (LDS broadcast reads) ----
    float best = -1e30f;
    int   barg = 0;
    for (int m = 0; m < M; ++m) {
      const float bx1 = s_ann[m * 5 + 0], by1 = s_ann[m * 5 + 1];
      const float bx2 = s_ann[m * 5 + 2], by2 = s_ann[m * 5 + 3];
      const float iw = fmaxf(fminf(ax2, bx2) - fmaxf(ax1, bx1), 0.0f);
      const float ih = fmaxf(fminf(ay2, by2) - fmaxf(ay1, by1), 0.0f);
      const float inter = iw * ih;
      const float ua = fmaxf(area_a + s_area[m] - inter, 1e-8f);
      float iou = inter / ua;
      iou = (s_valid[m] != 0.0f) ? iou : -1.0f;
      if (iou > best) { best = iou; barg = m; }   // strict '>' keeps first argmax
    }

    const bool pos = (best >= 0.5f);
    const bool neg = (best < 0.4f);

    // ---- classification focal loss (skip 'ignore' rows entirely) ----
    if (pos || neg) {
      const int label = pos ? (int)s_ann[barg * 5 + 4] : -1;
      const float* row = cls + ((size_t)b * A + a) * K;
      int k = 0;
      if ((K & 3) == 0) {
        const float4* row4 = (const float4*)row;
        // 4 independent accumulators: break the serial FMA chain so VOPD
        // dual-issue / trans-op co-execution can overlap with v_log_f32.
        float acc0 = 0.0f, acc1 = 0.0f, acc2 = 0.0f, acc3 = 0.0f;
#pragma unroll 4
        for (int j = 0; j < (K >> 2); ++j) {
          const float4 c4 = row4[j];
          const int kb = j * 4;
          const float c0 = fminf(fmaxf(c4.x, 1e-4f), 1.0f - 1e-4f);
          const float c1 = fminf(fmaxf(c4.y, 1e-4f), 1.0f - 1e-4f);
          const float c2 = fminf(fmaxf(c4.z, 1e-4f), 1.0f - 1e-4f);
          const float c3 = fminf(fmaxf(c4.w, 1e-4f), 1.0f - 1e-4f);
          acc0 += (kb + 0 == label)
              ? (ALPHA_C * (1.0f - c0) * (1.0f - c0) * (-__logf(c0)))
              : ((1.0f - ALPHA_C) * c0 * c0 * (-__logf(1.0f - c0)));
          acc1 += (kb + 1 == label)
              ? (ALPHA_C * (1.0f - c1) * (1.0f - c1) * (-__logf(c1)))
              : ((1.0f - ALPHA_C) * c1 * c1 * (-__logf(1.0f - c1)));
          acc2 += (kb + 2 == label)
              ? (ALPHA_C * (1.0f - c2) * (1.0f - c2) * (-__logf(c2)))
              : ((1.0f - ALPHA_C) * c2 * c2 * (-__logf(1.0f - c2)));
          acc3 += (kb + 3 == label)
              ? (ALPHA_C * (1.0f - c3) * (1.0f - c3) * (-__logf(c3)))
              : ((1.0f - ALPHA_C) * c3 * c3 * (-__logf(1.0f - c3)));
        }
        cls_acc = (acc0 + acc1) + (acc2 + acc3);
        k = K;
      }
      for (; k < K; ++k) {
        const float cc = fminf(fmaxf(row[k], 1e-4f), 1.0f - 1e-4f);
        cls_acc += (k == label)
            ? (ALPHA_C * (1.0f - cc) * (1.0f - cc) * (-__logf(cc)))
            : ((1.0f - ALPHA_C) * cc * cc * (-__logf(1.0f - cc)));
      }
    }

    // ---- smooth-L1 regression loss (positives only) ----
    if (pos) {
      pos_f = 1.0f;
      const float4 rg = *(const float4*)(regr + ((size_t)b * A + a) * 4);
      const float gx1 = s_ann[barg * 5 + 0], gy1 = s_ann[barg * 5 + 1];
      const float gx2 = s_ann[barg * 5 + 2], gy2 = s_ann[barg * 5 + 3];
      const float aw = ax2 - ax1, ah = ay2 - ay1;
      const float acx = ax1 + 0.5f * aw, acy = ay1 + 0.5f * ah;
      const float gwr = gx2 - gx1, ghr = gy2 - gy1;
      const float gcx = gx1 + 0.5f * gwr, gcy = gy1 + 0.5f * ghr;
      const float gw = fmaxf(gwr, 1.0f), gh = fmaxf(ghr, 1.0f);
      const float tt[4] = {
          ((gcx - acx) / aw) / 0.1f,
          ((gcy - acy) / ah) / 0.1f,
          __logf(gw / aw) / 0.2f,
          __logf(gh / ah) / 0.2f };
      const float rr[4] = {rg.x, rg.y, rg.z, rg.w};
#pragma unroll
      for (int i = 0; i < 4; ++i) {
        const float d = fabsf(tt[i] - rr[i]);
        reg_acc += (d <= (1.0f / 9.0f)) ? (4.5f * d * d) : (d - (0.5f / 9.0f));
      }
    }
  }

  // ---- deterministic block reduction, one partial triple per block ----
  const float c_tot = block_reduce_sum(cls_acc, s_red, tid);
  const float r_tot = block_reduce_sum(reg_acc, s_red, tid);
  const float p_tot = block_reduce_sum(pos_f,   s_red, tid);
  if (tid == 0) {
    const size_t idx = ((size_t)b * nblk + bx) * 3;
    part[idx + 0] = c_tot;
    part[idx + 1] = r_tot;
    part[idx + 2] = p_tot;
  }
}

__global__ __launch_bounds__(BLK)
void focal_finalize(const float* __restrict__ part,
                    const float* __restrict__ ann,
                    float* __restrict__ out,
                    int B, int M, int nblk) {
  __shared__ float s_red[BLK];
  const int tid = threadIdx.x;
  float cls_mean = 0.0f, reg_mean = 0.0f;
  for (int b = 0; b < B; ++b) {
    float c = 0.0f, r = 0.0f, p = 0.0f;
    for (int i = tid; i < nblk; i += BLK) {
      const size_t idx = ((size_t)b * nblk + i) * 3;
      c += part[idx + 0];
      r += part[idx + 1];
      p += part[idx + 2];
    }
    c = block_reduce_sum(c, s_red, tid);
    r = block_reduce_sum(r, s_red, tid);
    p = block_reduce_sum(p, s_red, tid);
    if (tid == 0) {
      bool hv = false;
      for (int m = 0; m < M; ++m)
        hv = hv || (ann[((size_t)b * M + m) * 5 + 4] != -1.0f);
      const float cls_t = hv ? (c / fmaxf(p, 1.0f)) : 0.0f;
      const float reg_t = (p > 0.0f) ? (r / fmaxf(p * 4.0f, 1.0f)) : 0.0f;
      cls_mean += cls_t;
      reg_mean += reg_t;
    }
  }
  if (tid == 0) {
    out[0] = cls_mean / (float)B;
    out[1] = reg_mean / (float)B;
  }
}

extern "C" void kernel_launch(void* const* d_in, const int* in_sizes, int n_in,
                              void* d_out, int out_size, void* d_ws, size_t ws_size,
                              hipStream_t stream) {
  (void)n_in; (void)out_size; (void)ws_size;
  const float* cls  = (const float*)d_in[0];   // [B,A,K]
  const float* regr = (const float*)d_in[1];   // [B,A,4]
  const float* anch = (const float*)d_in[2];   // [1,A,4]
  const float* ann  = (const float*)d_in[3];   // [B,M,5]

  const int A = in_sizes[2] / 4;
  const int B = in_sizes[1] / (A * 4);
  const int K = in_sizes[0] / (B * A);
  const int M = in_sizes[3] / (B * 5);
  const int nblk = (A + BLK - 1) / BLK;

  float* part = (float*)d_ws;                  // nblk*B*3 floats (~45 KB)

  dim3 grid(nblk, B);
  focal_main<<<grid, BLK, 0, stream>>>(cls, regr, anch, ann, part, A, K, M, nblk);
  focal_finalize<<<1, BLK, 0, stream>>>(part, ann, (float*)d_out, B, M, nblk);
}